// Nequix_26688926777500
// MI455X (gfx1250) — compile-verified
//
#include <hip/hip_runtime.h>
#include <hip/hip_bf16.h>

// ---------------------------------------------------------------------------
// Nequix layer for MI455X (gfx1250): fp32 WMMA 16x16x4 for all GEMMs,
// LDS-staged radial MLP with async (ASYNCcnt) double-buffered weight tiles,
// atomic scatter for segment_sum.
// ---------------------------------------------------------------------------

typedef __attribute__((ext_vector_type(2))) float v2f;
typedef __attribute__((ext_vector_type(8))) float v8f;

#define NN 4000
#define EE 128000
#define CC 128
#define HH 64

__device__ __forceinline__ v8f wmma4(v2f a, v2f b, v8f c) {
  // D = A(16x4) * B(4x16) + C(16x16), fp32.
  return __builtin_amdgcn_wmma_f32_16x16x4_f32(
      /*neg_a=*/false, a, /*neg_b=*/false, b,
      /*c_mod=*/(short)0, c, /*reuse_a=*/false, /*reuse_b=*/false);
}

__device__ __forceinline__ float silu_f(float x) {
  return x / (1.0f + __expf(-x));
}

// Async global->LDS 16B copy (GLOBAL_LOAD_ASYNC_TO_LDS_B128, ASYNCcnt).
__device__ __forceinline__ void async_cp16(void* lds_dst, const void* gsrc) {
  const unsigned int l = (unsigned int)(unsigned long long)lds_dst;
  asm volatile("global_load_async_to_lds_b128 %0, %1, off"
               :: "v"(l), "v"(gsrc) : "memory");
}
__device__ __forceinline__ void wait_async0() {
  asm volatile("s_wait_asynccnt 0x0" ::: "memory");
}

__constant__ float kINV_C  = 0.08838834764831845f;   // 1/sqrt(128)
__constant__ float kINV_2C = 0.0625f;                // 1/sqrt(256)
__constant__ float kINV_NB = 0.17677669529663689f;   // 1/sqrt(32)
__constant__ float kSQRT3  = 1.7320508075688772f;

// Generic 16-row GEMM accumulate: A rows from ap (element stride STR),
// B from W (leading dim ldb), one 16-col tile at jcol.
template <int STR>
__device__ __forceinline__ v8f gemm_acc(const float* __restrict__ ap,
                                        const float* __restrict__ W,
                                        int ldb, int K, int jcol, int half,
                                        v8f acc) {
  for (int k0 = 0; k0 < K; k0 += 4) {
    const int ka = k0 + 2 * half;
    v2f a, b;
    a.x = ap[ka * STR];
    a.y = ap[(ka + 1) * STR];
    b.x = W[ka * ldb + jcol];
    b.y = W[(ka + 1) * ldb + jcol];
    acc = wmma4(a, b, acc);
  }
  return acc;
}

// ---------------------------------------------------------------------------
__global__ void zero_f32(float* __restrict__ p, long n) {
  long i = (long)blockIdx.x * blockDim.x + threadIdx.x;
  long s = (long)gridDim.x * blockDim.x;
  for (; i < n; i += s) p[i] = 0.0f;
}

// ---------------------------------------------------------------------------
// y0 = xs @ W1_0 * inv_c   (g==0);  y1[:,:,i] = xv[:,:,i] @ W1_1 * inv_c
// grid: (N/16, 4), block: 128 (4 waves, 2 col-tiles each)
__global__ void node_linear1(const float* __restrict__ xs,
                             const float* __restrict__ xv,
                             const float* __restrict__ W1_0,
                             const float* __restrict__ W1_1,
                             float* __restrict__ y0,
                             float* __restrict__ y1) {
  const int lane = threadIdx.x & 31;
  const int wv   = threadIdx.x >> 5;
  const int row  = lane & 15;
  const int half = lane >> 4;
  const int col  = lane & 15;
  const int nb   = blockIdx.x * 16;
  const int g    = blockIdx.y;
  const float* W  = (g == 0) ? W1_0 : W1_1;
  const float* ap = (g == 0) ? xs + (long)(nb + row) * CC
                             : xv + (long)(nb + row) * CC * 3 + (g - 1);

  for (int jj = 0; jj < 2; ++jj) {
    const int j = wv * 2 + jj;
    v8f acc = {};
    if (g == 0) acc = gemm_acc<1>(ap, W, CC, CC, j * 16 + col, half, acc);
    else        acc = gemm_acc<3>(ap, W, CC, CC, j * 16 + col, half, acc);
    for (int v = 0; v < 8; ++v) {
      const int rg = nb + v + 8 * half;
      const int c  = j * 16 + col;
      const float val = acc[v] * kINV_C;
      if (g == 0) y0[rg * CC + c] = val;
      else        y1[(rg * CC + c) * 3 + (g - 1)] = val;
    }
  }
}

// ---------------------------------------------------------------------------
// Stage a 64x64 weight, transposed into 4 col tiles: Wt[j][c][k] = W[k*64+j*16+c]
__device__ __forceinline__ void stage_w64_t(float (*Wt)[16][68],
                                            const float* __restrict__ W,
                                            int tid) {
  for (int t = tid; t < 1024; t += 128) {
    const int r  = t >> 4;           // k row 0..63
    const int c4 = (t & 15) << 2;    // col 0,4,...,60
    const float4 v = *(const float4*)&W[r * 64 + c4];
    const int j  = c4 >> 4;
    const int cc = c4 & 15;
    Wt[j][cc + 0][r] = v.x;
    Wt[j][cc + 1][r] = v.y;
    Wt[j][cc + 2][r] = v.z;
    Wt[j][cc + 3][r] = v.w;
  }
}

// One 64->64 MLP layer with SiLU, per-wave 16-edge tile, h + weights in LDS.
__device__ __forceinline__ void mlp_layer64(float (*h)[68],
                                            const float (*Wt)[16][68],
                                            int row, int half, int col) {
  v8f acc[4];
  for (int j = 0; j < 4; ++j) acc[j] = (v8f){};
  for (int k0 = 0; k0 < HH; k0 += 4) {
    const int ka = k0 + 2 * half;
    const v2f a = *(const v2f*)&h[row][ka];
    for (int j = 0; j < 4; ++j) {
      const v2f b = *(const v2f*)&Wt[j][col][ka];
      acc[j] = wmma4(a, b, acc[j]);
    }
  }
  __syncthreads();   // all reads of old h done (uniform across block)
  for (int j = 0; j < 4; ++j)
    for (int v = 0; v < 8; ++v)
      h[v + 8 * half][j * 16 + col] = silu_f(acc[j][v]);
  __syncthreads();
}

// ---------------------------------------------------------------------------
// Fused: radial MLP (WMMA) + tensor product + scatter-add into agg_s/agg_v.
// block = 128 threads = 4 waves; wave handles 16 edges; grid = E/64.
__global__ void edge_kernel(const float* __restrict__ evec,
                            const float* __restrict__ rb,
                            const float* __restrict__ w0,
                            const float* __restrict__ w1,
                            const float* __restrict__ w2,
                            const float* __restrict__ w3,
                            const float* __restrict__ y0,
                            const float* __restrict__ y1,
                            const int* __restrict__ senders,
                            const int* __restrict__ receivers,
                            float* __restrict__ agg_s,
                            float* __restrict__ agg_v) {
  __shared__ float hsh[4][16][68];        // per-wave activations
  __shared__ float usm[4][16][3];         // unit vectors
  __shared__ int   ssm[4][16];            // senders
  __shared__ int   rsm[4][16];            // receivers
  __shared__ float w0s[4][8][16];         // layer0 weights, tiled
  __shared__ float w1t[4][16][68];        // layer1 weights, transposed tiles
  __shared__ float w2t[4][16][68];        // layer2 weights, transposed tiles
  __shared__ float w3s[2][64][16];        // layer3 tile, double-buffered

  const int tid  = threadIdx.x;
  const int lane = tid & 31;
  const int wv   = tid >> 5;
  const int row  = lane & 15;
  const int half = lane >> 4;
  const int col  = lane & 15;
  const int e0   = blockIdx.x * 64 + wv * 16;

  // --- cooperative weight staging -----------------------------------------
  for (int t = tid; t < 512; t += 128) {   // w0: w0s[j][k][c] = w0[k*64+j*16+c]
    const int k = t >> 6, c6 = t & 63;
    w0s[c6 >> 4][k][c6 & 15] = w0[t];
  }
  stage_w64_t(w1t, w1, tid);
  stage_w64_t(w2t, w2, tid);

  // --- per-edge metadata ---------------------------------------------------
  if (lane < 16) {
    const int e = e0 + lane;
    const float ex = evec[e * 3 + 0];
    const float ey = evec[e * 3 + 1];
    const float ez = evec[e * 3 + 2];
    const float r  = sqrtf(ex * ex + ey * ey + ez * ez);
    const float ir = 1.0f / fmaxf(r, 1e-12f);
    usm[wv][lane][0] = ex * ir;
    usm[wv][lane][1] = ey * ir;
    usm[wv][lane][2] = ez * ir;
    ssm[wv][lane] = senders[e];
    rsm[wv][lane] = receivers[e];
  }

  // kick off async copy of the first w3 tile (buffer 0)
  for (int t = tid; t < 256; t += 128) {
    const int r = t >> 2, c4 = (t & 3) << 2;
    async_cp16(&w3s[0][r][c4], &w3[r * 512 + c4]);
  }
  __syncthreads();   // staging + metadata visible

  float (*h)[68] = hsh[wv];

  // --- Layer 0: (16x8) @ (8x64) -> silu -> h ------------------------------
  for (int j = 0; j < 4; ++j) {
    v8f acc = {};
    for (int k0 = 0; k0 < 8; k0 += 4) {
      const int ka = k0 + 2 * half;
      const v2f a = *(const v2f*)&rb[(e0 + row) * 8 + ka];
      v2f b;
      b.x = w0s[j][ka][col];
      b.y = w0s[j][ka + 1][col];
      acc = wmma4(a, b, acc);
    }
    for (int v = 0; v < 8; ++v)
      h[v + 8 * half][j * 16 + col] = silu_f(acc[v]);
  }
  __syncthreads();

  // --- Layers 1, 2: 64 -> 64 ----------------------------------------------
  mlp_layer64(h, w1t, row, half, col);
  mlp_layer64(h, w2t, row, half, col);

  // --- Layer 3: (16x64) @ (64x512), double-buffered async weight tiles ----
#pragma unroll 1
  for (int j = 0; j < 32; ++j) {
    wait_async0();        // this wave's share of tile j landed in LDS
    __syncthreads();      // all waves' shares landed
    if (j + 1 < 32) {     // prefetch tile j+1 into the other buffer
      for (int t = tid; t < 256; t += 128) {
        const int r = t >> 2, c4 = (t & 3) << 2;
        async_cp16(&w3s[(j + 1) & 1][r][c4], &w3[r * 512 + (j + 1) * 16 + c4]);
      }
    }
    const float (*wt)[16] = w3s[j & 1];

    v8f acc = {};
    for (int k0 = 0; k0 < HH; k0 += 4) {
      const int ka = k0 + 2 * half;
      const v2f a = *(const v2f*)&h[row][ka];
      v2f b;
      b.x = wt[ka][col];
      b.y = wt[ka + 1][col];
      acc = wmma4(a, b, acc);
    }

    const int oc = j * 16 + col;   // 0..511, quarter is tile-uniform
    for (int v = 0; v < 8; ++v) {
      const int er  = v + 8 * half;
      const float wd = acc[v];
      const int snd = ssm[wv][er];
      const int rcv = rsm[wv][er];
      const float ux = usm[wv][er][0];
      const float uy = usm[wv][er][1];
      const float uz = usm[wv][er][2];
      if (oc < CC) {                       // scal part a: t0a*ws0
        const float m0v = y0[snd * CC + oc];
        atomicAdd(&agg_s[rcv * 256 + oc], m0v * wd * kINV_NB);
      } else if (oc < 2 * CC) {            // scal part b: t0b*ws1
        const int c = oc - CC;
        const float* m1p = &y1[(snd * CC + c) * 3];
        const float t0b = m1p[0] * ux + m1p[1] * uy + m1p[2] * uz;
        atomicAdd(&agg_s[rcv * 256 + CC + c], t0b * wd * kINV_NB);
      } else if (oc < 3 * CC) {            // vec part a: (m0*Y1)*wv0
        const int c = oc - 2 * CC;
        const float base = y0[snd * CC + c] * wd * kINV_NB * kSQRT3;
        atomicAdd(&agg_v[(rcv * 256 + c) * 3 + 0], base * ux);
        atomicAdd(&agg_v[(rcv * 256 + c) * 3 + 1], base * uy);
        atomicAdd(&agg_v[(rcv * 256 + c) * 3 + 2], base * uz);
      } else {                             // vec part b: m1*wv1
        const int c = oc - 3 * CC;
        const float* m1p = &y1[(snd * CC + c) * 3];
        const float t = wd * kINV_NB;
        atomicAdd(&agg_v[(rcv * 256 + CC + c) * 3 + 0], m1p[0] * t);
        atomicAdd(&agg_v[(rcv * 256 + CC + c) * 3 + 1], m1p[1] * t);
        atomicAdd(&agg_v[(rcv * 256 + CC + c) * 3 + 2], m1p[2] * t);
      }
    }
    __syncthreads();   // everyone done reading buffer (j&1) before re-fill
  }
}

// ---------------------------------------------------------------------------
// Output GEMMs (WMMA) + per-species skip (VALU) + SiLU gating.
// block = 256 threads (8 waves), 16 nodes per block, grid = N/16.
__global__ void node_out(const float* __restrict__ agg_s,
                         const float* __restrict__ agg_v,
                         const float* __restrict__ W2_0,
                         const float* __restrict__ W2_1,
                         const float* __restrict__ xs,
                         const float* __restrict__ xv,
                         const float* __restrict__ Wsk0,
                         const float* __restrict__ Wsk1,
                         const int* __restrict__ species,
                         float* __restrict__ out) {
  __shared__ float sbuf[16][256];
  __shared__ float vbuf[16][128][3];

  const int tid  = threadIdx.x;
  const int lane = tid & 31;
  const int wv   = tid >> 5;
  const int row  = lane & 15;
  const int half = lane >> 4;
  const int col  = lane & 15;
  const int nb   = blockIdx.x * 16;

  // 40 WMMA jobs: 16 tiles of s (256 cols), 24 tiles of v (3 comps x 8 tiles)
#pragma unroll 1
  for (int q = 0; q < 5; ++q) {
    const int jb = wv * 5 + q;
    if (jb < 16) {
      const int j = jb;
      v8f acc = {};
      acc = gemm_acc<1>(agg_s + (long)(nb + row) * 256, W2_0, 256, 256,
                        j * 16 + col, half, acc);
      for (int v = 0; v < 8; ++v)
        sbuf[v + 8 * half][j * 16 + col] = acc[v] * kINV_2C;
    } else {
      const int i = (jb - 16) / 8;
      const int j = (jb - 16) % 8;
      v8f acc = {};
      acc = gemm_acc<3>(agg_v + (long)(nb + row) * 256 * 3 + i, W2_1, CC, 256,
                        j * 16 + col, half, acc);
      for (int v = 0; v < 8; ++v)
        vbuf[v + 8 * half][j * 16 + col][i] = acc[v] * kINV_2C;
    }
  }
  __syncthreads();

  // Per-species skip connections (node-specific weight matrices).
#pragma unroll 1
  for (int n = 0; n < 16; ++n) {
    const int sp = species[nb + n];
    {
      float acc = 0.0f;
      const float* wp = Wsk0 + (long)sp * CC * 256 + tid;
      const float* xp = xs + (long)(nb + n) * CC;
      for (int c = 0; c < CC; ++c) acc += xp[c] * wp[c * 256];
      sbuf[n][tid] += acc * kINV_C;
    }
    for (int t = tid; t < 384; t += 256) {
      const int d = t / 3, i = t % 3;
      float acc = 0.0f;
      const float* wq = Wsk1 + (long)sp * CC * CC + d;
      const float* xq = xv + (long)(nb + n) * CC * 3 + i;
      for (int c = 0; c < CC; ++c) acc += xq[c * 3] * wq[c * CC];
      vbuf[n][d][i] += acc * kINV_C;
    }
  }
  __syncthreads();

  // Gating + output: out[n] = [silu(s[:C]) | (v * silu(s[C:])).reshape(3C)]
#pragma unroll 1
  for (int n = 0; n < 16; ++n) {
    float* o = out + (long)(nb + n) * 512;
    if (tid < CC) o[tid] = silu_f(sbuf[n][tid]);
    for (int t = tid; t < 384; t += 256) {
      const int c = t / 3, i = t % 3;
      o[CC + c * 3 + i] = vbuf[n][c][i] * silu_f(sbuf[n][CC + c]);
    }
  }
}

// ---------------------------------------------------------------------------
extern "C" void kernel_launch(void* const* d_in, const int* in_sizes, int n_in,
                              void* d_out, int out_size, void* d_ws, size_t ws_size,
                              hipStream_t stream) {
  const float* xs    = (const float*)d_in[0];
  const float* xv    = (const float*)d_in[1];
  const float* evec  = (const float*)d_in[2];
  const float* rb    = (const float*)d_in[3];
  const float* W1_0  = (const float*)d_in[4];
  const float* W1_1  = (const float*)d_in[5];
  const float* w0    = (const float*)d_in[6];
  const float* w1    = (const float*)d_in[7];
  const float* w2    = (const float*)d_in[8];
  const float* w3    = (const float*)d_in[9];
  const float* W2_0  = (const float*)d_in[10];
  const float* W2_1  = (const float*)d_in[11];
  const float* Wsk0  = (const float*)d_in[12];
  const float* Wsk1  = (const float*)d_in[13];
  const int* species   = (const int*)d_in[14];
  const int* senders   = (const int*)d_in[15];
  const int* receivers = (const int*)d_in[16];

  float* ws    = (float*)d_ws;
  float* y0    = ws;                 // N*C          = 512000
  float* y1    = ws + 512000;        // N*C*3        = 1536000
  float* agg_s = ws + 2048000;       // N*2C         = 1024000
  float* agg_v = ws + 3072000;       // N*2C*3       = 3072000
                                     // total        = 6144000 floats (24.6 MB)

  zero_f32<<<2048, 256, 0, stream>>>(agg_s, 4096000L);
  node_linear1<<<dim3(NN / 16, 4), 128, 0, stream>>>(xs, xv, W1_0, W1_1, y0, y1);
  edge_kernel<<<EE / 64, 128, 0, stream>>>(evec, rb, w0, w1, w2, w3, y0, y1,
                                           senders, receivers, agg_s, agg_v);
  node_out<<<NN / 16, 256, 0, stream>>>(agg_s, agg_v, W2_0, W2_1, xs, xv,
                                        Wsk0, Wsk1, species, (float*)d_out);
}